// RotatedRetinaNet_31250182045725
// MI455X (gfx1250) — compile-verified
//
#include <hip/hip_runtime.h>

#define NCLS 15
#define KSEL 512
#define SCORE_THR 0.05f
#define NMS_THR 0.5f
#define PI_F 3.14159265358979323846f

typedef float v2f __attribute__((ext_vector_type(2)));
typedef float v8f __attribute__((ext_vector_type(8)));

struct SelState { unsigned long long prefix; unsigned int need; unsigned int pad; };

// ---------------------------------------------------------------- init
__global__ void init_ws(unsigned int* __restrict__ hist, SelState* __restrict__ st,
                        unsigned int* __restrict__ cnt, int Bn) {
  int tid = blockIdx.x * blockDim.x + threadIdx.x;
  int total = Bn * 65536;
  if (tid < total) hist[tid] = 0u;
  if (tid < Bn) { st[tid].prefix = 0ull; st[tid].need = KSEL; st[tid].pad = 0u; cnt[tid] = 0u; }
}

// ------------------------------------------------- streaming score->key
// key = mono(max_logit) << 32 | (0xFFFFFFFF - idx)  (descending score, then ascending idx)
__global__ void score_keys(const float* __restrict__ logits,
                           unsigned long long* __restrict__ keys,
                           int Bn, int Nn) {
  long long tid = (long long)blockIdx.x * blockDim.x + threadIdx.x;
  long long total = (long long)Bn * Nn;
  if (tid >= total) return;
  const float* lg = logits + tid * NCLS;
  __builtin_prefetch(lg + NCLS * 256, 0, 1);   // global_prefetch_b8, stream ahead
  float m = lg[0];
#pragma unroll
  for (int c = 1; c < NCLS; ++c) { float v = lg[c]; m = v > m ? v : m; }
  unsigned int fb = __float_as_uint(m);
  fb = (fb & 0x80000000u) ? ~fb : (fb | 0x80000000u);   // order-preserving map
  unsigned int idx = (unsigned int)(tid % Nn);
  keys[tid] = ((unsigned long long)fb << 32) | (unsigned long long)(0xFFFFFFFFu - idx);
}

// --------------------------------------------------- radix-select passes
__global__ void hist_pass(const unsigned long long* __restrict__ keys,
                          unsigned int* __restrict__ hist,
                          const SelState* __restrict__ st, int Nn, int Bn, int level) {
  long long tid = (long long)blockIdx.x * blockDim.x + threadIdx.x;
  long long total = (long long)Bn * Nn;
  if (tid >= total) return;
  int b = (int)(tid / Nn);
  unsigned long long k = keys[tid];
  bool match = (level == 0) || ((k >> (64 - 16 * level)) == st[b].prefix);
  if (match) {
    unsigned int digit = (unsigned int)((k >> (48 - 16 * level)) & 0xFFFFull);
    atomicAdd(&hist[(size_t)b * 65536 + digit], 1u);
  }
}

__global__ void select_scan(unsigned int* __restrict__ hist, SelState* __restrict__ st) {
  int b = blockIdx.x;
  int t = threadIdx.x;                 // 256 threads
  __shared__ unsigned int part[256];
  unsigned int* h = hist + (size_t)b * 65536;
  unsigned int sum = 0;
  for (int k = 0; k < 256; ++k) sum += h[t * 256 + k];
  part[t] = sum;
  __syncthreads();
  if (t == 0) {
    unsigned int need = st[b].need;
    unsigned int cum = 0;
    int seg = 255;
    for (; seg > 0; --seg) {
      if (cum + part[seg] >= need) break;
      cum += part[seg];
    }
    int digit = seg * 256 + 255;
    for (; digit > seg * 256; --digit) {
      unsigned int c = h[digit];
      if (cum + c >= need) break;
      cum += c;
    }
    st[b].prefix = (st[b].prefix << 16) | (unsigned long long)(unsigned int)digit;
    st[b].need = need - cum;
  }
  __syncthreads();
  for (int k = 0; k < 256; ++k) h[t * 256 + k] = 0u;   // clear for next level
}

__global__ void compact_sel(const unsigned long long* __restrict__ keys,
                            const SelState* __restrict__ st,
                            unsigned int* __restrict__ cnt,
                            int* __restrict__ selIdx,
                            unsigned long long* __restrict__ selKey,
                            int Nn, int Bn) {
  long long tid = (long long)blockIdx.x * blockDim.x + threadIdx.x;
  long long total = (long long)Bn * Nn;
  if (tid >= total) return;
  int b = (int)(tid / Nn);
  unsigned long long k = keys[tid];
  if (k >= st[b].prefix) {           // pivot = exact 512th-largest unique key
    unsigned int slot = atomicAdd(&cnt[b], 1u);
    if (slot < KSEL) {
      selIdx[b * KSEL + slot] = (int)(tid % Nn);
      selKey[b * KSEL + slot] = k;
    }
  }
}

// ---------------------------------------------------- bitonic sort (desc)
__global__ void sort_sel(unsigned long long* __restrict__ selKey, int* __restrict__ selIdx) {
  int b = blockIdx.x, t = threadIdx.x;    // 512 threads
  __shared__ unsigned long long sk[KSEL];
  __shared__ int si[KSEL];
  sk[t] = selKey[b * KSEL + t];
  si[t] = selIdx[b * KSEL + t];
  __syncthreads();
  for (int k = 2; k <= KSEL; k <<= 1) {
    for (int j = k >> 1; j > 0; j >>= 1) {
      int ixj = t ^ j;
      if (ixj > t) {
        bool up = ((t & k) == 0);
        unsigned long long a = sk[t], c = sk[ixj];
        bool dosw = up ? (a < c) : (a > c);   // descending
        if (dosw) {
          sk[t] = c; sk[ixj] = a;
          int tmp = si[t]; si[t] = si[ixj]; si[ixj] = tmp;
        }
      }
      __syncthreads();
    }
  }
  selIdx[b * KSEL + t] = si[t];
  selKey[b * KSEL + t] = sk[t];
}

// ------------------------------------------- decode + outputs + geometry
__global__ void finalize_cand(const float* __restrict__ logits,
                              const float* __restrict__ deltas,
                              const float* __restrict__ anchors,
                              const int* __restrict__ selIdx,
                              float* __restrict__ out,
                              float* __restrict__ corners,
                              float* __restrict__ areaWH,
                              float4* __restrict__ cvec,
                              int* __restrict__ valid,
                              int Nn, int Bn) {
  int b = blockIdx.x, k = threadIdx.x;    // KSEL threads
  int idx = selIdx[b * KSEL + k];
  const float* lg = logits + ((long long)b * Nn + idx) * NCLS;
  float best = lg[0]; int lab = 0;
#pragma unroll
  for (int c = 1; c < NCLS; ++c) { float v = lg[c]; if (v > best) { best = v; lab = c; } }
  float score = 1.0f / (1.0f + expf(-best));
  const float* dl = deltas + ((long long)b * Nn + idx) * 5;
  const float* an = anchors + (long long)idx * 5;
  float ax = an[0], ay = an[1], aw = an[2], ah = an[3], aa = an[4];
  float px = ax + dl[0] * aw;
  float py = ay + dl[1] * ah;
  float pw = aw * expf(fminf(fmaxf(dl[2], -4.f), 4.f));
  float ph = ah * expf(fminf(fmaxf(dl[3], -4.f), 4.f));
  float pa = aa + dl[4] * (180.0f / PI_F);

  float* ob = out + ((long long)b * KSEL + k) * 5;
  ob[0] = px; ob[1] = py; ob[2] = pw; ob[3] = ph; ob[4] = pa;
  int sBase = Bn * KSEL * 5;
  out[sBase + b * KSEL + k] = score;
  out[sBase + Bn * KSEL + b * KSEL + k] = (float)lab;

  float th = pa * (PI_F / 180.0f);
  float cs = cosf(th), sn = sinf(th);
  float hx = pw * 0.5f, hy = ph * 0.5f;
  const float lx[4] = { -hx, hx, hx, -hx };
  const float ly[4] = { -hy, -hy, hy, hy };
  float* cr = corners + ((long long)b * KSEL + k) * 8;
#pragma unroll
  for (int v = 0; v < 4; ++v) {
    cr[2 * v]     = px + cs * lx[v] - sn * ly[v];
    cr[2 * v + 1] = py + sn * lx[v] + cs * ly[v];
  }
  areaWH[b * KSEL + k] = pw * ph;
  float r = 0.5f * sqrtf(pw * pw + ph * ph);
  cvec[b * KSEL + k] = make_float4(px, py, r, px * px + py * py - r * r);
  valid[b * KSEL + k] = (score > SCORE_THR) ? 1 : 0;
}

// ----------------------- WMMA circumscribed-circle overlap prefilter
// s_ij = dist^2 - (ri+rj)^2 = p_i + p_j - 2*dot(u_i,u_j), u=(cx,cy,r,0).
// One wave computes one 16x16 tile of the 512x512 Gram matrix with
// v_wmma_f32_16x16x4_f32. s>0 => provably disjoint => IoU = 0.
__global__ void __launch_bounds__(32) wmma_prefilter(const float4* __restrict__ cvec,
                                                     unsigned char* __restrict__ mask) {
  int bid = blockIdx.x;
  int b = bid >> 10;           // 1024 tiles per image (32x32)
  int tile = bid & 1023;
  int i0 = (tile >> 5) * 16, j0 = (tile & 31) * 16;
  int l = threadIdx.x;         // full wave32, EXEC all ones
  int lm = l & 15;
  int half = l >> 4;
  const float4* base = cvec + b * KSEL;
  float4 ci = base[i0 + lm];
  float4 cj = base[j0 + lm];
  // A 16x4 layout: VGPR0 = K0 (lanes0-15) / K2 (lanes16-31), VGPR1 = K1 / K3
  v2f a, bb;
  a.x  = half ? ci.z : ci.x;   a.y  = half ? 0.0f : ci.y;
  bb.x = half ? cj.z : cj.x;   bb.y = half ? 0.0f : cj.y;
  v8f c = {};
  v8f d = __builtin_amdgcn_wmma_f32_16x16x4_f32(false, a, false, bb, (short)0, c, false, false);
  float pj = cj.w;             // column n = lm
  unsigned char* mb = mask + (size_t)b * KSEL * KSEL;
#pragma unroll
  for (int v = 0; v < 8; ++v) {
    int mi = v + 8 * half;     // D layout: VGPR v -> row v (lanes0-15) / v+8 (lanes16-31)
    float pi = base[i0 + mi].w;
    float s = pi + pj - 2.0f * d[v];
    mb[(size_t)(i0 + mi) * KSEL + (j0 + lm)] = (s <= 0.0f) ? (unsigned char)1 : (unsigned char)0;
  }
}

// ---------------------------------- exact rotated IoU (Sutherland-Hodgman)
__global__ void iou_pairs(const float* __restrict__ corners,
                          const float* __restrict__ areaWH,
                          const unsigned char* __restrict__ mask,
                          float* __restrict__ iou, int Bn) {
  long long tid = (long long)blockIdx.x * blockDim.x + threadIdx.x;
  long long total = (long long)Bn * KSEL * KSEL;
  if (tid >= total) return;
  int b = (int)(tid / (KSEL * KSEL));
  int rem = (int)(tid % (KSEL * KSEL));
  int i = rem / KSEL, j = rem % KSEL;
  if (j <= i) { iou[tid] = 0.0f; return; }
  if (!mask[tid]) { iou[tid] = 0.0f; return; }   // WMMA prefilter: disjoint

  const float* ca = corners + ((long long)b * KSEL + i) * 8;
  const float* cb = corners + ((long long)b * KSEL + j) * 8;
  float xs[2][8], ys[2][8];
  int cur = 0, n = 4;
#pragma unroll
  for (int v = 0; v < 4; ++v) { xs[0][v] = ca[2 * v]; ys[0][v] = ca[2 * v + 1]; }
  for (int e = 0; e < 4; ++e) {
    int e1 = (e + 1) & 3;
    float axp = cb[2 * e], ayp = cb[2 * e + 1];
    float ex = cb[2 * e1] - axp, ey = cb[2 * e1 + 1] - ayp;
    int nxt = cur ^ 1, m = 0;
    for (int v = 0; v < n; ++v) {
      int vn = (v + 1 == n) ? 0 : v + 1;
      float dc = ex * (ys[cur][v] - ayp) - ey * (xs[cur][v] - axp);
      float dn = ex * (ys[cur][vn] - ayp) - ey * (xs[cur][vn] - axp);
      bool cin = dc >= 0.0f, nin = dn >= 0.0f;
      if (cin) { xs[nxt][m] = xs[cur][v]; ys[nxt][m] = ys[cur][v]; ++m; }
      if (cin != nin) {
        float denom = dc - dn;
        if (fabsf(denom) < 1e-9f) denom = 1e-9f;
        float t = dc / denom;
        xs[nxt][m] = xs[cur][v] + t * (xs[cur][vn] - xs[cur][v]);
        ys[nxt][m] = ys[cur][v] + t * (ys[cur][vn] - ys[cur][v]);
        ++m;
      }
    }
    cur = nxt; n = m;
    if (n == 0) break;
  }
  float inter = 0.0f;
  if (n >= 3) {
    float acc = 0.0f;
    for (int v = 0; v < n; ++v) {
      int vn = (v + 1 == n) ? 0 : v + 1;
      acc += xs[cur][v] * ys[cur][vn] - ys[cur][v] * xs[cur][vn];
    }
    inter = 0.5f * fabsf(acc);
  }
  float u = areaWH[b * KSEL + i] + areaWH[b * KSEL + j] - inter;
  iou[tid] = inter / fmaxf(u, 1e-6f);
}

// --------------------------------------------------------- greedy NMS
__global__ void nms_kernel(const float* __restrict__ iou,
                           const int* __restrict__ valid,
                           float* __restrict__ out, int Bn) {
  int b = blockIdx.x, j = threadIdx.x;      // KSEL threads
  __shared__ unsigned char keep[KSEL];
  keep[j] = (unsigned char)valid[b * KSEL + j];
  __syncthreads();
  const float* im = iou + (size_t)b * KSEL * KSEL;
  for (int i = 0; i < KSEL - 1; ++i) {
    bool ki = keep[i] != 0;                 // keep[i] never written at iteration i
    if (ki && j > i && im[(size_t)i * KSEL + j] > NMS_THR) keep[j] = 0;
    __syncthreads();
  }
  out[Bn * KSEL * 5 + 2 * Bn * KSEL + b * KSEL + j] = keep[j] ? 1.0f : 0.0f;
}

// ================================================================ launch
extern "C" void kernel_launch(void* const* d_in, const int* in_sizes, int n_in,
                              void* d_out, int out_size, void* d_ws, size_t ws_size,
                              hipStream_t stream) {
  const float* logits  = (const float*)d_in[0];  // (B,N,15)
  const float* deltas  = (const float*)d_in[1];  // (B,N,5)
  const float* anchors = (const float*)d_in[2];  // (N,5)
  float* out = (float*)d_out;

  const int Nn = in_sizes[2] / 5;
  const int Bn = in_sizes[1] / (Nn * 5);

  char* w = (char*)d_ws;
  size_t off = 0;
  auto take = [&](size_t bytes) -> char* {
    char* p = w + off;
    off += (bytes + 255) & ~(size_t)255;
    return p;
  };
  unsigned long long* keys  = (unsigned long long*)take((size_t)Bn * Nn * 8);
  unsigned int*       hist  = (unsigned int*)      take((size_t)Bn * 65536 * 4);
  SelState*           st    = (SelState*)          take((size_t)Bn * sizeof(SelState));
  unsigned int*       cnt   = (unsigned int*)      take((size_t)Bn * 4);
  int*                selIdx= (int*)               take((size_t)Bn * KSEL * 4);
  unsigned long long* selKey= (unsigned long long*)take((size_t)Bn * KSEL * 8);
  float*              crn   = (float*)             take((size_t)Bn * KSEL * 8 * 4);
  float*              areaWH= (float*)             take((size_t)Bn * KSEL * 4);
  float4*             cvec  = (float4*)            take((size_t)Bn * KSEL * 16);
  int*                valid = (int*)               take((size_t)Bn * KSEL * 4);
  unsigned char*      mask  = (unsigned char*)     take((size_t)Bn * KSEL * KSEL);
  float*              iouM  = (float*)             take((size_t)Bn * KSEL * KSEL * 4);
  (void)ws_size; (void)n_in; (void)out_size;

  const long long totalBN = (long long)Bn * Nn;
  const int bnBlocks = (int)((totalBN + 255) / 256);

  init_ws<<<(Bn * 65536 + 255) / 256, 256, 0, stream>>>(hist, st, cnt, Bn);
  score_keys<<<bnBlocks, 256, 0, stream>>>(logits, keys, Bn, Nn);
  for (int level = 0; level < 4; ++level) {
    hist_pass<<<bnBlocks, 256, 0, stream>>>(keys, hist, st, Nn, Bn, level);
    select_scan<<<Bn, 256, 0, stream>>>(hist, st);
  }
  compact_sel<<<bnBlocks, 256, 0, stream>>>(keys, st, cnt, selIdx, selKey, Nn, Bn);
  sort_sel<<<Bn, KSEL, 0, stream>>>(selKey, selIdx);
  finalize_cand<<<Bn, KSEL, 0, stream>>>(logits, deltas, anchors, selIdx, out,
                                         crn, areaWH, cvec, valid, Nn, Bn);
  wmma_prefilter<<<Bn * 1024, 32, 0, stream>>>(cvec, mask);
  iou_pairs<<<(int)(((long long)Bn * KSEL * KSEL + 255) / 256), 256, 0, stream>>>(
      crn, areaWH, mask, iouM, Bn);
  nms_kernel<<<Bn, KSEL, 0, stream>>>(iouM, valid, out, Bn);
}